// TrainableBridge_45801531244692
// MI455X (gfx1250) — compile-verified
//
#include <hip/hip_runtime.h>
#include <math.h>

// MI455X (gfx1250) implementation.
// Roofline: 2MB in + 32MB out mandatory HBM; intermediates (72MB) stay in the
// 192MB L2 -> whole pipeline is ~2-3us memory-floor. Bicubic resize is cast as
// 16x16 tile GEMMs (K=8, two V_WMMA_F32_16X16X4_F32 per tile) since a 16-row
// output tile's 4-tap windows span at most 8 input rows. Soft-dilate runs in
// exp-domain so the middle iteration needs no transcendentals.

typedef __attribute__((ext_vector_type(2))) float v2f;
typedef __attribute__((ext_vector_type(8))) float v8f;

#define H_IN  256
#define W_IN  256
#define H_OUT 1024
#define W_OUT 1024
#define NB    8
#define SC    (255.0f / 1023.0f)   // linspace(0, n_in-1, n_out) step

__device__ __forceinline__ int clampi(int v, int lo, int hi) {
    return v < lo ? lo : (v > hi ? hi : v);
}

// Keys cubic (a=-0.75) weights for output index o, scattered into the 8-slot
// K-window starting at `base` (clipped taps fold into their clipped slot).
__device__ __forceinline__ void taps8(int o, int base, float* wrow) {
    const float a = -0.75f;
    float src = (float)o * SC;
    float i0f = floorf(src);
    int   i0  = (int)i0f;
    float t   = src - i0f;
    float t1 = t + 1.0f;
    float w0 = ((a * t1 - 5.0f * a) * t1 + 8.0f * a) * t1 - 4.0f * a;
    float w1 = ((a + 2.0f) * t - (a + 3.0f)) * t * t + 1.0f;
    float u  = 1.0f - t;
    float w2 = ((a + 2.0f) * u - (a + 3.0f)) * u * u + 1.0f;
    float u2 = 2.0f - t;
    float w3 = ((a * u2 - 5.0f * a) * u2 + 8.0f * a) * u2 - 4.0f * a;
    float w[4] = {w0, w1, w2, w3};
    int idx[4];
#pragma unroll
    for (int j = 0; j < 4; ++j) idx[j] = clampi(i0 - 1 + j, 0, H_IN - 1);
#pragma unroll
    for (int k = 0; k < 8; ++k) {
        float acc = 0.0f;
#pragma unroll
        for (int j = 0; j < 4; ++j) acc += (idx[j] - base == k) ? w[j] : 0.0f;
        wrow[k] = acc;
    }
}

// ---------------- Vertical resize (sigmoid fused) : (8,256,256)->(8,1024,256)
// One wave = one 16x16 output tile. Y = W(16x8) x X(8x16), two K=4 WMMAs.
__global__ void vresize_wmma(const float* __restrict__ in, float* __restrict__ out) {
    const int l  = threadIdx.x;        // 0..31
    const int m  = l & 15;             // M (lanes 0-15) / M again (16-31)
    const int hi = l >> 4;             // half-wave select
    const int kb = hi * 2;             // K pair base: 0 or 2
    const int c0 = blockIdx.x * 16;    // col tile in [0,256)
    const int h0 = blockIdx.y * 16;    // output-row tile in [0,1024)
    const int b  = blockIdx.z;
    const int base = (int)floorf((float)h0 * SC) - 1;  // K-window start (can be -1)

    // A = weight matrix (16x8): lane holds row m, cols {kb,kb+1} / {kb+4,kb+5}
    float wrow[8];
    taps8(h0 + m, base, wrow);
    v2f A0, A1;
    A0.x = wrow[kb];     A0.y = wrow[kb + 1];
    A1.x = wrow[kb + 4]; A1.y = wrow[kb + 5];

    // B = sigmoid(input) (8x16): lane holds col m, rows base+{kb,kb+1,kb+4,kb+5}
    const float* src = in + (size_t)b * (H_IN * W_IN);
    const int col = c0 + m;
    int r0 = clampi(base + kb,     0, H_IN - 1);
    int r1 = clampi(base + kb + 1, 0, H_IN - 1);
    int r2 = clampi(base + kb + 4, 0, H_IN - 1);
    int r3 = clampi(base + kb + 5, 0, H_IN - 1);
    v2f B0, B1;
    B0.x = 1.0f / (1.0f + expf(-src[r0 * W_IN + col]));
    B0.y = 1.0f / (1.0f + expf(-src[r1 * W_IN + col]));
    B1.x = 1.0f / (1.0f + expf(-src[r2 * W_IN + col]));
    B1.y = 1.0f / (1.0f + expf(-src[r3 * W_IN + col]));

    v8f C = {};
    C = __builtin_amdgcn_wmma_f32_16x16x4_f32(false, A0, false, B0, (short)0, C, false, false);
    C = __builtin_amdgcn_wmma_f32_16x16x4_f32(false, A1, false, B1, (short)0, C, false, false);

    // D layout: vgpr j -> row j (lanes 0-15) / j+8 (lanes 16-31)
    float* dst = out + ((size_t)b * H_OUT + h0 + hi * 8) * W_IN + col;
#pragma unroll
    for (int j = 0; j < 8; ++j) dst[(size_t)j * W_IN] = C[j];
}

// ---------------- Horizontal resize : (8,1024,256)->(8,1024,1024)
// Y = X(16x8) x W'(8x16), two K=4 WMMAs per tile.
__global__ void hresize_wmma(const float* __restrict__ in, float* __restrict__ out) {
    const int l   = threadIdx.x;
    const int m   = l & 15;
    const int hi  = l >> 4;
    const int kb  = hi * 2;
    const int wt0 = blockIdx.x * 16;   // output-col tile
    const int h0  = blockIdx.y * 16;   // row tile
    const int b   = blockIdx.z;
    const int base = (int)floorf((float)wt0 * SC) - 1;

    // A = data (16x8): lane holds row h0+m, input cols base+{kb,kb+1,kb+4,kb+5}
    const float* src = in + ((size_t)b * H_OUT + h0 + m) * W_IN;
    v2f A0, A1;
    A0.x = src[clampi(base + kb,     0, W_IN - 1)];
    A0.y = src[clampi(base + kb + 1, 0, W_IN - 1)];
    A1.x = src[clampi(base + kb + 4, 0, W_IN - 1)];
    A1.y = src[clampi(base + kb + 5, 0, W_IN - 1)];

    // B = weights (8x16): lane holds output col wt0+m
    float wcol[8];
    taps8(wt0 + m, base, wcol);
    v2f B0, B1;
    B0.x = wcol[kb];     B0.y = wcol[kb + 1];
    B1.x = wcol[kb + 4]; B1.y = wcol[kb + 5];

    v8f C = {};
    C = __builtin_amdgcn_wmma_f32_16x16x4_f32(false, A0, false, B0, (short)0, C, false, false);
    C = __builtin_amdgcn_wmma_f32_16x16x4_f32(false, A1, false, B1, (short)0, C, false, false);

    float* dst = out + ((size_t)b * H_OUT + h0 + hi * 8) * W_OUT + wt0 + m;
#pragma unroll
    for (int j = 0; j < 8; ++j) dst[(size_t)j * W_OUT] = C[j];
}

// ---------------- 3x3 Gaussian (zero pad) fused with e0 = exp(5*(x-1))
__global__ void smooth_exp_kernel(const float* __restrict__ in, float* __restrict__ out) {
    const int x = blockIdx.x * blockDim.x + threadIdx.x;
    const int y = blockIdx.y;
    const int b = blockIdx.z;
    const float g1 = 0.60653066f;                 // exp(-0.5), sigma = 1
    const float wc = 1.0f / (1.0f + 2.0f * g1);   // center 1D weight
    const float we = g1 * wc;                     // edge 1D weight
    const float wv[3] = {we, wc, we};
    const float* p = in + (size_t)b * H_OUT * W_OUT;
    float acc = 0.0f;
#pragma unroll
    for (int dy = -1; dy <= 1; ++dy) {
        int yy = y + dy;
        bool vy = (yy >= 0) && (yy < H_OUT);
#pragma unroll
        for (int dx = -1; dx <= 1; ++dx) {
            int xx = x + dx;
            bool v = vy && (xx >= 0) && (xx < W_OUT);
            float val = v ? p[(size_t)yy * W_OUT + xx] : 0.0f;
            acc += val * (wv[dy + 1] * wv[dx + 1]);
        }
    }
    out[((size_t)b * H_OUT + y) * W_OUT + x] = expf(5.0f * (acc - 1.0f));
}

// ---------------- Soft dilate in exp-domain.
// e' = exp(5*(clip(1+log(s)/5,0,1)-1)) == clamp(s, e^-5, 1); pad value = e^-5.
template <bool FINAL>
__global__ void dilate_kernel(const float* __restrict__ in, float* __restrict__ out) {
    const int x = blockIdx.x * blockDim.x + threadIdx.x;
    const int y = blockIdx.y;
    const int b = blockIdx.z;
    const float EPAD = 0.006737946999085467f;     // exp(-5) == exp(beta*(0-1))
    const float* p = in + (size_t)b * H_OUT * W_OUT;
    float s = 0.0f;
#pragma unroll
    for (int dy = -1; dy <= 1; ++dy) {
        int yy = y + dy;
        bool vy = (yy >= 0) && (yy < H_OUT);
#pragma unroll
        for (int dx = -1; dx <= 1; ++dx) {
            int xx = x + dx;
            bool v = vy && (xx >= 0) && (xx < W_OUT);
            s += v ? p[(size_t)yy * W_OUT + xx] : EPAD;
        }
    }
    float r;
    if (FINAL) {
        r = 1.0f + 0.2f * logf(s);                // 1 + log(s)/beta
        r = fminf(fmaxf(r, 0.0f), 1.0f);
    } else {
        r = fminf(fmaxf(s, EPAD), 1.0f);          // stays in exp-domain
    }
    out[((size_t)b * H_OUT + y) * W_OUT + x] = r;
}

extern "C" void kernel_launch(void* const* d_in, const int* in_sizes, int n_in,
                              void* d_out, int out_size, void* d_ws, size_t ws_size,
                              hipStream_t stream) {
    const float* in = (const float*)d_in[0];
    float* out = (float*)d_out;

    // Workspace layout (72 MB total): ws0 = 8x1024x256, ws1/ws2 = 8x1024x1024
    float* ws0 = (float*)d_ws;
    float* ws1 = ws0 + (size_t)NB * H_OUT * W_IN;
    float* ws2 = ws1 + (size_t)NB * H_OUT * W_OUT;

    dim3 gv(W_IN / 16, H_OUT / 16, NB);      // 16 x 64 x 8 one-wave tiles
    vresize_wmma<<<gv, 32, 0, stream>>>(in, ws0);

    dim3 gh(W_OUT / 16, H_OUT / 16, NB);     // 64 x 64 x 8 one-wave tiles
    hresize_wmma<<<gh, 32, 0, stream>>>(ws0, ws1);

    dim3 gp(W_OUT / 256, H_OUT, NB);
    smooth_exp_kernel<<<gp, 256, 0, stream>>>(ws1, ws2);
    dilate_kernel<false><<<gp, 256, 0, stream>>>(ws2, ws1);
    dilate_kernel<true><<<gp, 256, 0, stream>>>(ws1, out);
}